// GCN_19215683682381
// MI455X (gfx1250) — compile-verified
//
#include <hip/hip_runtime.h>
#include <hip/hip_bf16.h>

// GCN forward for MI455X (gfx1250, wave32).
// Dominant cost = edge gather/scatter (~3.3 GB, L2-resident: h and agg are
// 25.6 MB each, global L2 is 192 MB). GEMMs are ~3.3 GFLOP -> negligible, so
// we keep them in full fp32 using native CDNA5 V_WMMA_F32_16X16X4_F32.

typedef __attribute__((ext_vector_type(2))) float v2f;
typedef __attribute__((ext_vector_type(8))) float v8f;

#define FEAT     128
#define THREADS  256
#define NGRAPHS  128
#define NCLASSES 10

__device__ __forceinline__ void atomic_add_f32(float* p, float v) {
  // non-returning global_atomic_add_f32, device scope
  (void)__hip_atomic_fetch_add(p, v, __ATOMIC_RELAXED, __HIP_MEMORY_SCOPE_AGENT);
}

// ---------------- utility: zero fill ----------------
__global__ void k_zero(float* __restrict__ p, int n) {
  int i = blockIdx.x * blockDim.x + threadIdx.x;
  if (i < n) p[i] = 0.0f;
}

__global__ void k_zero4(float4* __restrict__ p, int n4) {
  int i = blockIdx.x * blockDim.x + threadIdx.x;
  if (i < n4) p[i] = make_float4(0.f, 0.f, 0.f, 0.f);
}

// ---------------- degree / dinv ----------------
__global__ void k_deg(const int* __restrict__ col, const float* __restrict__ ew,
                      float* __restrict__ deg, int n_edges) {
  int e = blockIdx.x * blockDim.x + threadIdx.x;
  if (e < n_edges) atomic_add_f32(&deg[col[e]], ew[e]);
}

__global__ void k_dinv(float* __restrict__ deg, int n) {
  int i = blockIdx.x * blockDim.x + threadIdx.x;
  if (i < n) deg[i] = rsqrtf(deg[i] + 1.0f);  // +1 = self-loop weight; always > 0
}

// ---------------- fp32 WMMA GEMM:  Y[n,128] = X[n,128] @ W[128,128] ----------------
// One wave computes a 16-row x 128-col strip via 8 accumulators, K in 32 steps of 4.
// A (16x4 f32): lane m = row, hi half holds K=2,3 -> v2f {K, K+1} with K = 4*step + 2*hi.
// B (4x16 f32): lane m = col, same K split -> v2f from LDS-resident W^T.
// C/D (16x16 f32): v8f, element j at row = 8*hi + j, col = lane&15.
__global__ __launch_bounds__(THREADS)
void k_gemm_wmma(const float* __restrict__ X, const float* __restrict__ W,
                 float* __restrict__ Y, int n_rows) {
  // W transposed into LDS with XOR bank swizzle (keeps 64 KB, kills the
  // 16-way conflict of a naive 512 B row stride). Swizzle XORs bits [5:1] of k.
  __shared__ float Wt[FEAT * FEAT];
  for (int idx = threadIdx.x; idx < FEAT * FEAT; idx += THREADS) {
    const int k = idx >> 7;        // row of W
    const int c = idx & 127;       // col of W
    Wt[c * FEAT + (k ^ ((c & 31) << 1))] = W[idx];
  }
  __syncthreads();

  const int lane = threadIdx.x & 31;
  const int wave = threadIdx.x >> 5;          // 8 waves / block
  const int m    = lane & 15;
  const int hi   = lane >> 4;                 // 0 or 1
  const int row0 = (blockIdx.x * 8 + wave) * 16;
  if (row0 >= n_rows) return;

  int rA = row0 + m;
  if (rA >= n_rows) rA = n_rows - 1;          // clamp (stores are guarded)
  const float* __restrict__ xrow = X + (size_t)rA * FEAT;

  v8f acc[8] = {};

  for (int step = 0; step < 32; ++step) {
    const int k0 = step * 4 + hi * 2;
    v2f a;
    a.x = xrow[k0];
    a.y = xrow[k0 + 1];
#pragma unroll
    for (int nt = 0; nt < 8; ++nt) {
      const int col  = nt * 16 + m;
      const int base = col * FEAT + (k0 ^ ((col & 31) << 1));
      v2f b;
      b.x = Wt[base];
      b.y = Wt[base + 1];
      // D = A*B + C  (8-arg form: neg_a, A, neg_b, B, c_mod, C, reuse_a, reuse_b)
      acc[nt] = __builtin_amdgcn_wmma_f32_16x16x4_f32(
          false, a, false, b, (short)0, acc[nt], false, false);
    }
  }

#pragma unroll
  for (int nt = 0; nt < 8; ++nt) {
#pragma unroll
    for (int j = 0; j < 8; ++j) {
      const int r = row0 + hi * 8 + j;
      if (r < n_rows) Y[(size_t)r * FEAT + nt * 16 + m] = acc[nt][j];
    }
  }
}

// ---------------- edge scatter: agg[col] += norm * h[row] ----------------
// One wave per edge: lane handles 4 contiguous features (512 B coalesced gather,
// 128 contiguous fp32 atomics). h and agg are L2-resident (25.6 MB each).
__global__ __launch_bounds__(THREADS)
void k_scatter(const int* __restrict__ rows, const int* __restrict__ cols,
               const float* __restrict__ ew, const float* __restrict__ dinv,
               const float* __restrict__ h, float* __restrict__ agg, int n_edges) {
  const long long t = (long long)blockIdx.x * THREADS + threadIdx.x;
  const int e = (int)(t >> 5);
  if (e >= n_edges) return;
  const int q = ((int)t & 31) << 2;
  const int r = rows[e];
  const int c = cols[e];
  const float nrm = dinv[r] * ew[e] * dinv[c];
  const float4 hv = *(const float4*)(h + (size_t)r * FEAT + q);
  float* dst = agg + (size_t)c * FEAT + q;
  atomic_add_f32(dst + 0, nrm * hv.x);
  atomic_add_f32(dst + 1, nrm * hv.y);
  atomic_add_f32(dst + 2, nrm * hv.z);
  atomic_add_f32(dst + 3, nrm * hv.w);
}

// ---------------- finish: act = relu(agg + dinv^2 * h + bias) (in place on agg) ----
__global__ void k_finish(float* __restrict__ agg, const float* __restrict__ h,
                         const float* __restrict__ dinv, const float* __restrict__ bias,
                         int total) {
  int i = blockIdx.x * blockDim.x + threadIdx.x;
  if (i >= total) return;
  const int node = i >> 7;
  const int f    = i & 127;
  const float di = dinv[node];
  const float v  = agg[i] + di * di * h[i] + bias[f];
  agg[i] = fmaxf(v, 0.0f);
}

// ---------------- global add pool into pooled[128,128] ----------------
__global__ void k_pool(const float* __restrict__ act, const int* __restrict__ batch,
                       float* __restrict__ pooled, int total) {
  int i = blockIdx.x * blockDim.x + threadIdx.x;
  if (i >= total) return;
  const int node = i >> 7;
  const int f    = i & 127;
  atomic_add_f32(&pooled[(size_t)batch[node] * FEAT + f], act[i]);
}

// ---------------- head: logits = pooled @ Wfc + bfc; log_softmax ----------------
__global__ __launch_bounds__(NGRAPHS)
void k_head(const float* __restrict__ pooled, const float* __restrict__ Wfc,
            const float* __restrict__ bfc, float* __restrict__ out) {
  const int g = threadIdx.x;  // one thread per graph
  float lg[NCLASSES];
#pragma unroll
  for (int c = 0; c < NCLASSES; ++c) lg[c] = bfc[c];
  for (int k = 0; k < FEAT; ++k) {
    const float p = pooled[g * FEAT + k];
#pragma unroll
    for (int c = 0; c < NCLASSES; ++c) lg[c] += p * Wfc[k * NCLASSES + c];
  }
  float mx = lg[0];
#pragma unroll
  for (int c = 1; c < NCLASSES; ++c) mx = fmaxf(mx, lg[c]);
  float s = 0.f;
#pragma unroll
  for (int c = 0; c < NCLASSES; ++c) s += expf(lg[c] - mx);
  const float lse = logf(s);
#pragma unroll
  for (int c = 0; c < NCLASSES; ++c) out[g * NCLASSES + c] = lg[c] - mx - lse;
}

extern "C" void kernel_launch(void* const* d_in, const int* in_sizes, int n_in,
                              void* d_out, int out_size, void* d_ws, size_t ws_size,
                              hipStream_t stream) {
  (void)n_in; (void)out_size; (void)ws_size;
  const float* x     = (const float*)d_in[0];
  const int*   ei    = (const int*)d_in[1];
  const int*   batch = (const int*)d_in[2];
  const float* ew    = (const float*)d_in[3];
  const float* W1    = (const float*)d_in[4];
  const float* b1    = (const float*)d_in[5];
  const float* W2    = (const float*)d_in[6];
  const float* b2    = (const float*)d_in[7];
  const float* Wfc   = (const float*)d_in[8];
  const float* bfc   = (const float*)d_in[9];
  float* out = (float*)d_out;

  const int N = in_sizes[0] / FEAT;   // 50000
  const int E = in_sizes[3];          // 1600000
  const int* rows = ei;               // edge_index[0]
  const int* cols = ei + E;           // edge_index[1]

  // workspace layout (floats): dinv | bufA (N*128) | bufB (N*128) | pooled
  float* ws = (float*)d_ws;
  size_t off = ((size_t)N + 255) & ~(size_t)255;
  float* dinv   = ws;
  float* bufA   = ws + off;            off += (size_t)N * FEAT;
  float* bufB   = ws + off;            off += (size_t)N * FEAT;
  float* pooled = ws + off;            // 128*128 floats

  const int nf        = N * FEAT;
  const int blk_n     = (N + THREADS - 1) / THREADS;
  const int blk_nf    = (nf + THREADS - 1) / THREADS;
  const int blk_nf4   = (nf / 4 + THREADS - 1) / THREADS;
  const int blk_e     = (E + THREADS - 1) / THREADS;
  const long long sct = (long long)E * 32;
  const int blk_sc    = (int)((sct + THREADS - 1) / THREADS);
  const int tiles     = (N + 15) / 16;
  const int blk_gemm  = (tiles + 7) / 8;

  // degree normalization
  k_zero<<<blk_n, THREADS, 0, stream>>>(dinv, N);
  k_deg <<<blk_e, THREADS, 0, stream>>>(cols, ew, dinv, E);
  k_dinv<<<blk_n, THREADS, 0, stream>>>(dinv, N);

  // layer 1: h = x@W1 -> bufA ; aggregate -> bufB ; act = relu(...) in bufB
  k_gemm_wmma<<<blk_gemm, THREADS, 0, stream>>>(x, W1, bufA, N);
  k_zero4    <<<blk_nf4,  THREADS, 0, stream>>>((float4*)bufB, nf / 4);
  k_scatter  <<<blk_sc,   THREADS, 0, stream>>>(rows, cols, ew, dinv, bufA, bufB, E);
  k_finish   <<<blk_nf,   THREADS, 0, stream>>>(bufB, bufA, dinv, b1, nf);

  // layer 2: h2 = act1@W2 -> bufA ; aggregate -> bufB ; act2 in bufB
  k_gemm_wmma<<<blk_gemm, THREADS, 0, stream>>>(bufB, W2, bufA, N);
  k_zero4    <<<blk_nf4,  THREADS, 0, stream>>>((float4*)bufB, nf / 4);
  k_scatter  <<<blk_sc,   THREADS, 0, stream>>>(rows, cols, ew, dinv, bufA, bufB, E);
  k_finish   <<<blk_nf,   THREADS, 0, stream>>>(bufB, bufA, dinv, b2, nf);

  // pool + head
  k_zero<<<(NGRAPHS * FEAT + THREADS - 1) / THREADS, THREADS, 0, stream>>>(pooled, NGRAPHS * FEAT);
  k_pool<<<blk_nf, THREADS, 0, stream>>>(bufB, batch, pooled, nf);
  k_head<<<1, NGRAPHS, 0, stream>>>(pooled, Wfc, bfc, out);
}